// Self_Attention_Global_11141145166181
// MI455X (gfx1250) — compile-verified
//
#include <hip/hip_runtime.h>
#include <math.h>

typedef float v2f __attribute__((ext_vector_type(2)));
typedef float v8f __attribute__((ext_vector_type(8)));

#define B_ 2
#define N_ 16384
#define C_ 64
#define H_ 8
#define NSLICE 32
#define ROWS_PER_SLICE (N_ / NSLICE)   // 512

// workspace layout (float offsets)
#define WS_GPART 0                              // [B][NSLICE][64*64] partial Gram
#define WS_PC    (B_ * NSLICE * 4096)           // [B][H][64*64] per-head P contribution
#define WS_P     (WS_PC + B_ * H_ * 4096)       // [B][64*64] folded projection matrix
// total = 335872 floats = ~1.31 MB

// ---------------------------------------------------------------------------
// Kernel 1: partial Gram matrices  Gpart[b][slice] = emb_slice^T @ emb_slice
// 16 waves per block, one 16x16 tile of G each; K-steps of 4 rows via WMMA f32.
// ---------------------------------------------------------------------------
__global__ __launch_bounds__(512)
void g_partial_kernel(const float* __restrict__ emb, float* __restrict__ ws) {
    const int b     = blockIdx.x / NSLICE;
    const int slice = blockIdx.x % NSLICE;
    const int wave  = threadIdx.x >> 5;
    const int lane  = threadIdx.x & 31;
    const int ctile = wave >> 2;        // M tile of G (c index)
    const int ftile = wave & 3;         // N tile of G (f index)
    const int rlo   = lane & 15;        // M (for A) / N (for B) within tile
    const int khalf = (lane >> 4) * 2;  // K pair selector

    const float* eb = emb + (size_t)b * N_ * C_;
    const int n0 = slice * ROWS_PER_SLICE;

    v8f acc = {0.f, 0.f, 0.f, 0.f, 0.f, 0.f, 0.f, 0.f};
    for (int s = 0; s < ROWS_PER_SLICE; s += 4) {
        const float* rowp = eb + (size_t)(n0 + s + khalf) * C_;
        v2f a, bb;
        a.x  = rowp[ctile * 16 + rlo];        // A[m,k]   = emb[n, c]  (emb^T)
        a.y  = rowp[C_ + ctile * 16 + rlo];   // next K (next row of emb)
        bb.x = rowp[ftile * 16 + rlo];        // B[k,n']  = emb[n, f]
        bb.y = rowp[C_ + ftile * 16 + rlo];
        acc = __builtin_amdgcn_wmma_f32_16x16x4_f32(
            false, a, false, bb, (short)0, acc, false, false);
    }

    float* gp = ws + WS_GPART + (size_t)(b * NSLICE + slice) * 4096;
    const int rbase = ctile * 16 + ((lane >> 4) << 3);  // lanes 16-31 hold M+8
#pragma unroll
    for (int r = 0; r < 8; ++r)
        gp[(rbase + r) * 64 + ftile * 16 + rlo] = acc[r];
}

// ---------------------------------------------------------------------------
// Kernel 2: per (b,h): S = Wq_h G Wk_h^T ; instance-norm ; softmax(rows) ;
//           M = attn @ Wv_h ; Pcontrib = Wo_h @ M.   All 64x64, done in LDS.
// ---------------------------------------------------------------------------
__global__ __launch_bounds__(256)
void heads_kernel(const float* __restrict__ Wq, const float* __restrict__ Wk,
                  const float* __restrict__ Wv, const float* __restrict__ Wo,
                  float* __restrict__ ws) {
    __shared__ float sG[4096], sW[4096], sT[4096], sS[4096];
    __shared__ float red[256], red2[256];
    __shared__ float s_mean, s_rstd;

    const int b = blockIdx.x >> 3;
    const int h = blockIdx.x & 7;
    const int tid = threadIdx.x;

    // deterministic reduce of Gram partials
    for (int i = tid; i < 4096; i += 256) {
        const float* gp = ws + WS_GPART + (size_t)b * NSLICE * 4096 + i;
        float s = 0.f;
        for (int sl = 0; sl < NSLICE; ++sl) s += gp[sl * 4096];
        sG[i] = s;
    }
    for (int i = tid; i < 4096; i += 256) sW[i] = Wk[h * 4096 + i];
    __syncthreads();

    // T1[e,d] = sum_f G[e,f] * Wk_h[d,f]
    {
        const int row = tid >> 2, c0 = (tid & 3) * 16;
        for (int j = 0; j < 16; ++j) {
            const int col = c0 + j;
            float s = 0.f;
            for (int f = 0; f < 64; ++f) s += sG[row * 64 + f] * sW[col * 64 + f];
            sT[row * 64 + col] = s;
        }
    }
    __syncthreads();
    for (int i = tid; i < 4096; i += 256) sW[i] = Wq[h * 4096 + i];
    __syncthreads();

    // S[c,d] = sum_e Wq_h[c,e] * T1[e,d]
    {
        const int row = tid >> 2, c0 = (tid & 3) * 16;
        for (int j = 0; j < 16; ++j) {
            const int col = c0 + j;
            float s = 0.f;
            for (int e = 0; e < 64; ++e) s += sW[row * 64 + e] * sT[e * 64 + col];
            sS[row * 64 + col] = s;
        }
    }
    __syncthreads();

    // InstanceNorm over the full 64x64 map (biased variance, eps=1e-5)
    {
        float s = 0.f, s2 = 0.f;
        for (int i = tid; i < 4096; i += 256) { float x = sS[i]; s += x; s2 += x * x; }
        red[tid] = s; red2[tid] = s2;
        __syncthreads();
        for (int off = 128; off > 0; off >>= 1) {
            if (tid < off) { red[tid] += red[tid + off]; red2[tid] += red2[tid + off]; }
            __syncthreads();
        }
        if (tid == 0) {
            float mean = red[0] * (1.0f / 4096.0f);
            float var  = red2[0] * (1.0f / 4096.0f) - mean * mean;
            s_mean = mean;
            s_rstd = rsqrtf(var + 1e-5f);
        }
        __syncthreads();
    }
    for (int i = tid; i < 4096; i += 256) sS[i] = (sS[i] - s_mean) * s_rstd;
    __syncthreads();

    // softmax along d (each of threads 0..63 owns one row)
    if (tid < 64) {
        const int r = tid * 64;
        float m = sS[r];
        for (int d = 1; d < 64; ++d) m = fmaxf(m, sS[r + d]);
        float s = 0.f;
        for (int d = 0; d < 64; ++d) { float e = expf(sS[r + d] - m); sS[r + d] = e; s += e; }
        const float inv = 1.f / s;
        for (int d = 0; d < 64; ++d) sS[r + d] *= inv;
    }
    __syncthreads();
    for (int i = tid; i < 4096; i += 256) sW[i] = Wv[h * 4096 + i];
    __syncthreads();

    // M[c,e] = sum_d attn[c,d] * Wv_h[d,e]  -> sT
    {
        const int row = tid >> 2, c0 = (tid & 3) * 16;
        for (int j = 0; j < 16; ++j) {
            const int col = c0 + j;
            float s = 0.f;
            for (int d = 0; d < 64; ++d) s += sS[row * 64 + d] * sW[d * 64 + col];
            sT[row * 64 + col] = s;
        }
    }
    __syncthreads();
    // Wo_h[c',c] = Wo[c'*512 + c*8 + h]
    for (int i = tid; i < 4096; i += 256) {
        const int cp = i >> 6, c = i & 63;
        sW[i] = Wo[cp * 512 + c * 8 + h];
    }
    __syncthreads();

    // Pcontrib[c',e] = sum_c Wo_h[c',c] * M[c,e]
    {
        float* pc = ws + WS_PC + (size_t)(b * H_ + h) * 4096;
        const int row = tid >> 2, c0 = (tid & 3) * 16;
        for (int j = 0; j < 16; ++j) {
            const int col = c0 + j;
            float s = 0.f;
            for (int c = 0; c < 64; ++c) s += sW[row * 64 + c] * sT[c * 64 + col];
            pc[row * 64 + col] = s;
        }
    }
}

// ---------------------------------------------------------------------------
// Kernel 3: P[b] = sum_h Pcontrib[b][h]   (deterministic)
// ---------------------------------------------------------------------------
__global__ __launch_bounds__(256)
void p_reduce_kernel(float* __restrict__ ws) {
    const int i = blockIdx.x * 256 + threadIdx.x;  // 0..8191
    const int b = i >> 12, j = i & 4095;
    const float* pc = ws + WS_PC + (size_t)b * H_ * 4096 + j;
    float s = 0.f;
    for (int h = 0; h < H_; ++h) s += pc[h * 4096];
    ws[WS_P + i] = s;
}

// ---------------------------------------------------------------------------
// Kernel 4: out_b[n,c'] = emb_b[n,:] . P_b[c',:]   (WMMA, P staged in LDS)
// Each wave: one 16-row stripe x all 64 output columns (4 accumulators).
// ---------------------------------------------------------------------------
__global__ __launch_bounds__(256)
void out_kernel(const float* __restrict__ emb, const float* __restrict__ ws,
                float* __restrict__ out) {
    __shared__ float sP[4096];
    const int batch = blockIdx.x >> 7;   // 128 blocks per batch
    const int blk   = blockIdx.x & 127;
    const int tid   = threadIdx.x;

    const float* P = ws + WS_P + (size_t)batch * 4096;
    for (int i = tid; i < 4096; i += 256) sP[i] = P[i];
    __syncthreads();

    const int wave = tid >> 5, lane = tid & 31;
    const int n0   = (blk * 8 + wave) * 16;
    const int rlo  = lane & 15;
    const int kh   = (lane >> 4) * 2;

    const float* A = emb + ((size_t)batch * N_ + n0) * C_;

    v8f acc0 = {0.f,0.f,0.f,0.f,0.f,0.f,0.f,0.f};
    v8f acc1 = acc0, acc2 = acc0, acc3 = acc0;

#pragma unroll
    for (int k = 0; k < 16; ++k) {
        v2f a  = *(const v2f*)(A + rlo * C_ + k * 4 + kh);            // emb rows
        v2f b0 = *(const v2f*)(&sP[( 0 + rlo) * 64 + k * 4 + kh]);    // P[c',e] = B[k,n']
        v2f b1 = *(const v2f*)(&sP[(16 + rlo) * 64 + k * 4 + kh]);
        v2f b2 = *(const v2f*)(&sP[(32 + rlo) * 64 + k * 4 + kh]);
        v2f b3 = *(const v2f*)(&sP[(48 + rlo) * 64 + k * 4 + kh]);
        acc0 = __builtin_amdgcn_wmma_f32_16x16x4_f32(false, a, false, b0, (short)0, acc0, false, false);
        acc1 = __builtin_amdgcn_wmma_f32_16x16x4_f32(false, a, false, b1, (short)0, acc1, false, false);
        acc2 = __builtin_amdgcn_wmma_f32_16x16x4_f32(false, a, false, b2, (short)0, acc2, false, false);
        acc3 = __builtin_amdgcn_wmma_f32_16x16x4_f32(false, a, false, b3, (short)0, acc3, false, false);
    }

    float* O = out + ((size_t)batch * N_ + n0) * C_;
    const int rbase = (lane >> 4) * 8;
#pragma unroll
    for (int r = 0; r < 8; ++r) {
        O[(rbase + r) * C_ +  0 + rlo] = acc0[r];
        O[(rbase + r) * C_ + 16 + rlo] = acc1[r];
        O[(rbase + r) * C_ + 32 + rlo] = acc2[r];
        O[(rbase + r) * C_ + 48 + rlo] = acc3[r];
    }
}

// ---------------------------------------------------------------------------
extern "C" void kernel_launch(void* const* d_in, const int* in_sizes, int n_in,
                              void* d_out, int out_size, void* d_ws, size_t ws_size,
                              hipStream_t stream) {
    const float* emb = (const float*)d_in[0];
    const float* Wq  = (const float*)d_in[1];
    const float* Wk  = (const float*)d_in[2];
    const float* Wv  = (const float*)d_in[3];
    const float* Wo  = (const float*)d_in[4];
    float* ws  = (float*)d_ws;
    float* out = (float*)d_out;

    hipLaunchKernelGGL(g_partial_kernel, dim3(B_ * NSLICE), dim3(512), 0, stream, emb, ws);
    hipLaunchKernelGGL(heads_kernel,     dim3(B_ * H_),     dim3(256), 0, stream, Wq, Wk, Wv, Wo, ws);
    hipLaunchKernelGGL(p_reduce_kernel,  dim3(32),          dim3(256), 0, stream, ws);
    hipLaunchKernelGGL(out_kernel,       dim3(256),         dim3(256), 0, stream, emb, ws, out);
}